// BiLSTMModel_81535659148061
// MI455X (gfx1250) — compile-verified
//
#include <hip/hip_runtime.h>

// ---------------- model dims ----------------
#define NSEQ 512      // 2B sequences
#define TLEN 256      // L
#define EDIM 128      // embedding
#define HDIM 128      // lstm hidden
#define G4H  512      // 4H gate width
#define OHD  1024     // output hidden
#define NB   256      // pair batch

typedef __attribute__((ext_vector_type(2))) float v2f;
typedef __attribute__((ext_vector_type(8))) float v8f;
typedef __attribute__((ext_vector_type(4))) int   v4i;

#define AS1 __attribute__((address_space(1)))
#define AS3 __attribute__((address_space(3)))

typedef AS1 v4i* g4p;   // global-memory b128 pointer
typedef AS3 v4i* l4p;   // LDS b128 pointer

#if __has_builtin(__builtin_amdgcn_global_load_async_to_lds_b128)
#define USE_ASYNC_ZX 1
#else
#define USE_ASYNC_ZX 0
#endif

__device__ __forceinline__ float sigm_(float x) { return 1.0f / (1.0f + __expf(-x)); }

// D = A(16x4 f32) * B(4x16 f32) + C(16x16 f32), wave32 WMMA
__device__ __forceinline__ v8f wmma_f32(v2f a, v2f b, v8f c) {
    return __builtin_amdgcn_wmma_f32_16x16x4_f32(false, a, false, b, (short)0, c, false, false);
}

#if USE_ASYNC_ZX
__device__ __forceinline__ void async_ld_b128(const float* g, float* l) {
    float* gg = const_cast<float*>(g);
    __builtin_amdgcn_global_load_async_to_lds_b128((g4p)gg, (l4p)l, 0, 0);
}
__device__ __forceinline__ void wait_async0() {
#if __has_builtin(__builtin_amdgcn_s_wait_asynccnt)
    __builtin_amdgcn_s_wait_asynccnt(0);
#else
    asm volatile("s_wait_asynccnt 0x0" ::: "memory");
#endif
}
#endif

// ------------------------------------------------------------------
// lens[n] = count_nonzero of row n of inp (row 2b = s1[b], 2b+1 = s2[b])
// ------------------------------------------------------------------
__global__ void lens_kernel(const int* __restrict__ s1, const int* __restrict__ s2,
                            int* __restrict__ lens) {
    int n = blockIdx.x;
    const int* row = (n & 1) ? (s2 + (n >> 1) * TLEN) : (s1 + (n >> 1) * TLEN);
    int t = threadIdx.x;
    __shared__ int red[256];
    red[t] = (row[t] != 0) ? 1 : 0;
    __syncthreads();
    for (int s = 128; s > 0; s >>= 1) {
        if (t < s) red[t] += red[t + s];
        __syncthreads();
    }
    if (t == 0) lens[n] = red[0];
}

// ------------------------------------------------------------------
// Zx[dir][tok][0:512] = emb_W[token] @ Wx_dir   (fused gather + GEMM)
// grid: 1024 blocks (512/dir), 256 thr. Each block: 256 tokens, full N=512.
// LDS: Wx row-pair-interleaved 128x512 f32 (256KB) + emb stage 16x132
// ------------------------------------------------------------------
__global__ void zx_kernel(const int* __restrict__ s1, const int* __restrict__ s2,
                          const float* __restrict__ emb,
                          const float* __restrict__ Wf, const float* __restrict__ Wb,
                          float* __restrict__ zxf, float* __restrict__ zxb) {
    extern __shared__ float smem[];
    float* WxL  = smem;                 // 128*512, pair-interleaved
    float* embL = smem + 128 * 512;     // 16*132

    const int dir = blockIdx.x >> 9;
    const int blk = blockIdx.x & 511;
    const float* W = dir ? Wb : Wf;     // rows 0..127 = Wx part
    float* zx = dir ? zxb : zxf;
    const int tid = threadIdx.x;

    // pair-interleave: W[k][col] -> WxL[(k/2)*1024 + 2*col + (k&1)]
    for (int i = tid; i < 128 * 512; i += 256) {
        int k = i >> 9, col = i & 511;
        WxL[((k >> 1) << 10) + (col << 1) + (k & 1)] = W[i];
    }
    __syncthreads();

    const int wave = tid >> 5, lane = tid & 31;
    const int m16 = lane & 15, kh = (lane >> 4) * 2;
    const int colbase = wave * 64;

    for (int mt = 0; mt < 16; ++mt) {
        const int tokbase = blk * 256 + mt * 16;
        {   // stage 16 embedding rows (gather; table is L2-resident)
            int r  = tid >> 4;
            int c0 = (tid & 15) * 8;
            int tokidx = tokbase + r;
            int n = tokidx >> 8, t = tokidx & 255;
            int tok = (n & 1) ? s2[(n >> 1) * TLEN + t] : s1[(n >> 1) * TLEN + t];
            const float* er = emb + (long long)tok * EDIM + c0;
            float* dl = embL + r * 132 + c0;
#pragma unroll
            for (int k = 0; k < 8; ++k) dl[k] = er[k];
        }
        __syncthreads();

        v8f acc0 = {}, acc1 = {}, acc2 = {}, acc3 = {};
        for (int kb = 0; kb < 128; kb += 4) {
            v2f a = *(const v2f*)(embL + m16 * 132 + kb + kh);
            const float* bp = WxL + (((kb >> 1) + (kh >> 1)) << 10);
            v2f b0 = *(const v2f*)(bp + ((colbase      + m16) << 1));
            v2f b1 = *(const v2f*)(bp + ((colbase + 16 + m16) << 1));
            v2f b2 = *(const v2f*)(bp + ((colbase + 32 + m16) << 1));
            v2f b3 = *(const v2f*)(bp + ((colbase + 48 + m16) << 1));
            acc0 = wmma_f32(a, b0, acc0);
            acc1 = wmma_f32(a, b1, acc1);
            acc2 = wmma_f32(a, b2, acc2);
            acc3 = wmma_f32(a, b3, acc3);
        }
        const int rowoff = (lane >> 4) * 8;
#pragma unroll
        for (int v = 0; v < 8; ++v) {
            int rbase = (tokbase + v + rowoff) * G4H;
            zx[rbase + colbase +  0 + m16] = acc0[v];
            zx[rbase + colbase + 16 + m16] = acc1[v];
            zx[rbase + colbase + 32 + m16] = acc2[v];
            zx[rbase + colbase + 48 + m16] = acc3[v];
        }
        __syncthreads();
    }
}

// ------------------------------------------------------------------
// Persistent recurrence: 64 WGs (32 fwd + 32 bwd), 16 seqs each, all 256 steps.
// LDS: Wh pair-interleaved 128x512 (256KB) + h 16x132 + z/zx panel 16x512.
// Step-t Zx slice is async-loaded to LDS *before* the GEMM (overlap),
// each wave fetching exactly the 16x64 panel it consumes -> per-wave
// s_wait_asynccnt 0, no extra barrier. c-state and Sum(h) in registers.
// ------------------------------------------------------------------
__global__ void lstm_kernel(const float* __restrict__ Wf, const float* __restrict__ Wb,
                            const float* __restrict__ bf, const float* __restrict__ bb,
                            const float* __restrict__ zxf, const float* __restrict__ zxb,
                            const int* __restrict__ lens, float* __restrict__ seqvec) {
    extern __shared__ float smem[];
    float* WhL = smem;                   // 128*512, pair-interleaved
    float* hL  = smem + 128 * 512;       // 16*132
    float* zB  = hL + 16 * 132;          // 16*512 : zx landing + z exchange
    __shared__ int lensL[16];

    const int dir = blockIdx.x >> 5;
    const int sb  = blockIdx.x & 31;
    const int n0  = sb * 16;
    const float* W    = (dir ? Wb : Wf) + 128 * G4H;  // Wh rows of [E+H, 4H]
    const float* bias = dir ? bb : bf;
    const float* zx   = dir ? zxb : zxf;
    const int tid = threadIdx.x;

    for (int i = tid; i < 128 * 512; i += 256) {
        int k = i >> 9, col = i & 511;
        WhL[((k >> 1) << 10) + (col << 1) + (k & 1)] = W[i];
    }
    for (int i = tid; i < 16 * 132; i += 256) hL[i] = 0.0f;
    if (tid < 16) lensL[tid] = lens[n0 + tid];
    __syncthreads();

    const int wave = tid >> 5, lane = tid & 31;
    const int m16 = lane & 15, kh = (lane >> 4) * 2;
    const int colbase = wave * 64;
    const int rowoff  = (lane >> 4) * 8;

    // cell-phase mapping: one seq, 8 units per thread
    const int cseq = tid >> 4;
    const int u0   = (tid & 15) * 8;
    const int slen = lensL[cseq];
    float creg[8], sreg[8], bi[8], bj[8], bff[8], bo[8];
#pragma unroll
    for (int k = 0; k < 8; ++k) {
        creg[k] = 0.f; sreg[k] = 0.f;
        bi[k]  = bias[u0 + k];        bj[k] = bias[128 + u0 + k];
        bff[k] = bias[256 + u0 + k];  bo[k] = bias[384 + u0 + k];
    }

    for (int t = 0; t < TLEN; ++t) {
#if USE_ASYNC_ZX
        // ---- kick off async fetch of this step's Zx panel (overlaps GEMM) ----
#pragma unroll
        for (int k8 = 0; k8 < 8; ++k8) {
            int c    = lane + k8 * 32;          // 256 b128 chunks per wave panel
            int s    = c >> 4;                  // row 0..15
            int col4 = colbase + (c & 15) * 4;  // this wave's 64 columns
            int sl   = lensL[s];
            int te   = dir ? ((t < sl) ? (sl - 1 - t) : t) : t;
            const float* g = zx + ((n0 + s) * TLEN + te) * G4H + col4;
            async_ld_b128(g, zB + s * 512 + col4);
        }
#endif
        // ---- z = h @ Wh (M=16, N=512, K=128) ----
        v8f acc0 = {}, acc1 = {}, acc2 = {}, acc3 = {};
        for (int kb = 0; kb < 128; kb += 4) {
            v2f a = *(const v2f*)(hL + m16 * 132 + kb + kh);
            const float* bp = WhL + (((kb >> 1) + (kh >> 1)) << 10);
            v2f b0 = *(const v2f*)(bp + ((colbase      + m16) << 1));
            v2f b1 = *(const v2f*)(bp + ((colbase + 16 + m16) << 1));
            v2f b2 = *(const v2f*)(bp + ((colbase + 32 + m16) << 1));
            v2f b3 = *(const v2f*)(bp + ((colbase + 48 + m16) << 1));
            acc0 = wmma_f32(a, b0, acc0);
            acc1 = wmma_f32(a, b1, acc1);
            acc2 = wmma_f32(a, b2, acc2);
            acc3 = wmma_f32(a, b3, acc3);
        }
#if USE_ASYNC_ZX
        // ---- fold in Zx (already in LDS), overwrite panel with z ----
        wait_async0();
#pragma unroll
        for (int v = 0; v < 8; ++v) {
            int s = v + rowoff;
            int o = s * 512 + colbase + m16;
            zB[o +  0] = acc0[v] + zB[o +  0];
            zB[o + 16] = acc1[v] + zB[o + 16];
            zB[o + 32] = acc2[v] + zB[o + 32];
            zB[o + 48] = acc3[v] + zB[o + 48];
        }
#else
        // ---- fallback: synchronous Zx reads + prefetch of next step ----
#pragma unroll
        for (int v = 0; v < 8; ++v) {
            int s  = v + rowoff;
            int sl = lensL[s];
            int te = dir ? ((t < sl) ? (sl - 1 - t) : t) : t;
            int rbase = ((n0 + s) * TLEN + te) * G4H;
            zB[s * 512 + colbase +  0 + m16] = acc0[v] + zx[rbase + colbase +  0 + m16];
            zB[s * 512 + colbase + 16 + m16] = acc1[v] + zx[rbase + colbase + 16 + m16];
            zB[s * 512 + colbase + 32 + m16] = acc2[v] + zx[rbase + colbase + 32 + m16];
            zB[s * 512 + colbase + 48 + m16] = acc3[v] + zx[rbase + colbase + 48 + m16];
            if (t + 1 < TLEN) {
                int t2  = t + 1;
                int te2 = dir ? ((t2 < sl) ? (sl - 1 - t2) : t2) : t2;
                int rb2 = ((n0 + s) * TLEN + te2) * G4H;
                __builtin_prefetch(&zx[rb2 + colbase + m16], 0, 0);
            }
        }
#endif
        __syncthreads();
        // ---- LSTM cell (forget_bias = 1.0), masked update + running mean ----
        if (t < slen) {
            const float* zr = zB + cseq * 512;
#pragma unroll
            for (int k = 0; k < 8; ++k) {
                int u = u0 + k;
                float zi = zr[u]       + bi[k];
                float zj = zr[128 + u] + bj[k];
                float zf = zr[256 + u] + bff[k];
                float zo = zr[384 + u] + bo[k];
                float nc = creg[k] * sigm_(zf + 1.0f) + sigm_(zi) * tanhf(zj);
                float nh = tanhf(nc) * sigm_(zo);
                creg[k] = nc;
                sreg[k] += nh;
                hL[cseq * 132 + u] = nh;
            }
        }
        __syncthreads();
    }
#pragma unroll
    for (int k = 0; k < 8; ++k)
        seqvec[(n0 + cseq) * 256 + dir * 128 + u0 + k] = sreg[k] * (1.0f / 256.0f);
}

// ------------------------------------------------------------------
// hid = relu(feats @ W_mid + b_mid), feats = seq_vec viewed as [256,512]
// ------------------------------------------------------------------
__global__ void mid_kernel(const float* __restrict__ feats, const float* __restrict__ Wm,
                           const float* __restrict__ bm, float* __restrict__ hid) {
    __shared__ float fL[16 * 516];
    const int mt = blockIdx.x & 15;
    const int nq = blockIdx.x >> 4;
    const int tid = threadIdx.x;
    for (int i = tid; i < (16 * 512) / 4; i += 256) {
        int r = i >> 7, c = i & 127;
        ((float4*)(fL + r * 516))[c] = ((const float4*)(feats + (mt * 16 + r) * 512))[c];
    }
    __syncthreads();
    const int wave = tid >> 5, lane = tid & 31;
    const int m16 = lane & 15, kh = (lane >> 4) * 2;
    const int col0 = nq * 256 + wave * 32;
    v8f acc0 = {}, acc1 = {};
    for (int kb = 0; kb < 512; kb += 4) {
        v2f a = *(const v2f*)(fL + m16 * 516 + kb + kh);
        const float* br = Wm + (kb + kh) * OHD;
        v2f b0, b1;
        b0.x = br[col0 + m16];       b0.y = br[OHD + col0 + m16];
        b1.x = br[col0 + 16 + m16];  b1.y = br[OHD + col0 + 16 + m16];
        acc0 = wmma_f32(a, b0, acc0);
        acc1 = wmma_f32(a, b1, acc1);
    }
    const int rowoff = (lane >> 4) * 8;
#pragma unroll
    for (int v = 0; v < 8; ++v) {
        int r = mt * 16 + v + rowoff;
        float x0 = acc0[v] + bm[col0 + m16];
        float x1 = acc1[v] + bm[col0 + 16 + m16];
        hid[r * OHD + col0 + m16]      = x0 > 0.f ? x0 : 0.f;
        hid[r * OHD + col0 + 16 + m16] = x1 > 0.f ? x1 : 0.f;
    }
}

// out[b] = sigmoid(hid[b,:] . W_out + b_out)
__global__ void out_kernel(const float* __restrict__ hid, const float* __restrict__ Wo,
                           const float* __restrict__ bo, float* __restrict__ out) {
    int b = threadIdx.x;
    const float* hr = hid + b * OHD;
    float s = 0.f;
    for (int k = 0; k < OHD; ++k) s += hr[k] * Wo[k];
    out[b] = 1.0f / (1.0f + __expf(-(s + bo[0])));
}

// ------------------------------------------------------------------
extern "C" void kernel_launch(void* const* d_in, const int* in_sizes, int n_in,
                              void* d_out, int out_size, void* d_ws, size_t ws_size,
                              hipStream_t stream) {
    const int*   s1   = (const int*)d_in[0];
    const int*   s2   = (const int*)d_in[1];
    const float* embW = (const float*)d_in[2];
    const float* Wf   = (const float*)d_in[3];
    const float* bf   = (const float*)d_in[4];
    const float* Wb   = (const float*)d_in[5];
    const float* bb   = (const float*)d_in[6];
    const float* Wm   = (const float*)d_in[7];
    const float* bm   = (const float*)d_in[8];
    const float* Wo   = (const float*)d_in[9];
    const float* bo   = (const float*)d_in[10];
    float* out = (float*)d_out;

    // workspace layout
    float* zxf    = (float*)d_ws;                              // 512*256*512 f32
    float* zxb    = zxf + (size_t)NSEQ * TLEN * G4H;           // 512*256*512 f32
    int*   lens   = (int*)(zxb + (size_t)NSEQ * TLEN * G4H);   // 512 i32
    float* seqvec = (float*)(lens + NSEQ);                     // 512*256 f32
    float* hid    = seqvec + NSEQ * 256;                       // 256*1024 f32

    const int ldsZx   = (128 * 512 + 16 * 132) * 4;            // 270,592 B
    const int ldsLstm = (128 * 512 + 16 * 132 + 16 * 512) * 4; // 303,360 B
    (void)hipFuncSetAttribute((const void*)zx_kernel,
                              hipFuncAttributeMaxDynamicSharedMemorySize, ldsZx);
    (void)hipFuncSetAttribute((const void*)lstm_kernel,
                              hipFuncAttributeMaxDynamicSharedMemorySize, ldsLstm);

    lens_kernel<<<NSEQ, 256, 0, stream>>>(s1, s2, lens);
    zx_kernel<<<1024, 256, ldsZx, stream>>>(s1, s2, embW, Wf, Wb, zxf, zxb);
    lstm_kernel<<<64, 256, ldsLstm, stream>>>(Wf, Wb, bf, bb, zxf, zxb, lens, seqvec);
    mid_kernel<<<64, 256, 0, stream>>>(seqvec, Wm, bm, hid);
    out_kernel<<<1, 256, 0, stream>>>(hid, Wo, bo, out);
}